// MulliganNet_64776696758510
// MI455X (gfx1250) — compile-verified
//
#include <hip/hip_runtime.h>
#include <hip/hip_bf16.h>

#define B_TOT   65536
#define EDIM    32
#define HAND    7
#define DECK    60
#define CDIM    68

typedef __attribute__((ext_vector_type(2))) float v2f;
typedef __attribute__((ext_vector_type(8))) float v8f;

// ---------------------------------------------------------------------------
// Kernel 1: per-batch attention + pooling.  One wave32 per batch element.
// lane == feature index (E=32 == wave32).  Writes combined[b][68] to ws.
// ---------------------------------------------------------------------------
__global__ __launch_bounds__(256) void k_attn_pool(
    const float* __restrict__ feats,
    const int*   __restrict__ hand_ids,
    const int*   __restrict__ deck_ids,
    const float* __restrict__ embed,
    const float* __restrict__ in_proj_w,  // [96][32]
    const float* __restrict__ in_proj_b,  // [96]
    const float* __restrict__ out_proj_w, // [32][32]
    const float* __restrict__ out_proj_b, // [32]
    const float* __restrict__ ln_g,
    const float* __restrict__ ln_b,
    const float* __restrict__ deck_w,     // [32][32]
    const float* __restrict__ deck_b,     // [32]
    float*       __restrict__ combined)   // [B][68]
{
    // weights staged transposed: [e][f] so lane==f reads are bank-conflict-free
    __shared__ float s_qkvT[EDIM * 96];
    __shared__ float s_opT [EDIM * EDIM];
    __shared__ float s_dwT [EDIM * EDIM];
    __shared__ float s_ipb[96], s_opb[EDIM], s_lng[EDIM], s_lnb[EDIM], s_db[EDIM];

    const int tid = threadIdx.x;
    for (int i = tid; i < 96 * EDIM; i += 256) {
        int f = i / EDIM, e = i % EDIM;
        s_qkvT[e * 96 + f] = in_proj_w[i];
    }
    for (int i = tid; i < EDIM * EDIM; i += 256) {
        int f = i / EDIM, e = i % EDIM;
        s_opT[e * EDIM + f] = out_proj_w[i];
        s_dwT[e * EDIM + f] = deck_w[i];
    }
    if (tid < 96) s_ipb[tid] = in_proj_b[tid];
    if (tid < EDIM) {
        s_opb[tid] = out_proj_b[tid];
        s_lng[tid] = ln_g[tid];
        s_lnb[tid] = ln_b[tid];
        s_db [tid] = deck_b[tid];
    }
    __syncthreads();

    const int wave = tid >> 5;
    const int lane = tid & 31;
    const int b    = blockIdx.x * 8 + wave;

    // ---- hand embedding gather (L2-resident table) ----
    float he[HAND];
    bool  pad[HAND];
    #pragma unroll
    for (int t = 0; t < HAND; ++t) {
        int id = hand_ids[b * HAND + t];
        pad[t] = (id == 0);
        he[t]  = embed[id * EDIM + lane];
    }

    // ---- qkv projection: lane = output feature ----
    float q[HAND], k[HAND], v[HAND];
    #pragma unroll
    for (int t = 0; t < HAND; ++t) {
        float aq = s_ipb[lane], ak = s_ipb[32 + lane], av = s_ipb[64 + lane];
        #pragma unroll
        for (int e = 0; e < EDIM; ++e) {
            float x = __shfl(he[t], e, 32);
            aq = fmaf(x, s_qkvT[e * 96 + lane],      aq);
            ak = fmaf(x, s_qkvT[e * 96 + 32 + lane], ak);
            av = fmaf(x, s_qkvT[e * 96 + 64 + lane], av);
        }
        q[t] = aq; k[t] = ak; v[t] = av;
    }

    // ---- attention: head h = lane>>3, per-head dot via 8-lane xor reduce ----
    const float inv_sqrt_hd = 0.35355339059327373f;
    float o[HAND];
    #pragma unroll
    for (int i = 0; i < HAND; ++i) {
        float s[HAND];
        #pragma unroll
        for (int j = 0; j < HAND; ++j) {
            float p = q[i] * k[j];
            p += __shfl_xor(p, 1, 32);
            p += __shfl_xor(p, 2, 32);
            p += __shfl_xor(p, 4, 32);
            s[j] = pad[j] ? -__builtin_inff() : p * inv_sqrt_hd;
        }
        float m = s[0];
        #pragma unroll
        for (int j = 1; j < HAND; ++j) m = fmaxf(m, s[j]);
        float den = 0.f;
        #pragma unroll
        for (int j = 0; j < HAND; ++j) { s[j] = __expf(s[j] - m); den += s[j]; }
        float inv = 1.f / den;
        float acc = 0.f;
        #pragma unroll
        for (int j = 0; j < HAND; ++j) acc = fmaf(s[j], v[j], acc);
        o[i] = acc * inv;   // o[i][lane] since f == lane
    }

    // ---- out_proj + residual + LayerNorm + masked mean-pool ----
    float hpool = 0.f, cnt = 0.f;
    #pragma unroll
    for (int t = 0; t < HAND; ++t) {
        float a = s_opb[lane];
        #pragma unroll
        for (int e = 0; e < EDIM; ++e) {
            float x = __shfl(o[t], e, 32);
            a = fmaf(x, s_opT[e * EDIM + lane], a);
        }
        float hv = a + he[t];
        float mu = hv;
        #pragma unroll
        for (int off = 1; off < 32; off <<= 1) mu += __shfl_xor(mu, off, 32);
        mu *= (1.0f / 32.0f);
        float d = hv - mu;
        float var = d * d;
        #pragma unroll
        for (int off = 1; off < 32; off <<= 1) var += __shfl_xor(var, off, 32);
        var *= (1.0f / 32.0f);
        float nrm = d * rsqrtf(var + 1e-5f) * s_lng[lane] + s_lnb[lane];
        if (!pad[t]) { hpool += nrm; cnt += 1.f; }
    }
    hpool /= (cnt + 1e-8f);

    // ---- deck pooling + deck proj ----
    float dp = 0.f, dcnt = 0.f;
    for (int t = 0; t < DECK; ++t) {
        int id = deck_ids[b * DECK + t];
        if (id != 0) { dp += embed[id * EDIM + lane]; dcnt += 1.f; }
    }
    dp /= (dcnt + 1e-8f);
    float df = s_db[lane];
    #pragma unroll
    for (int e = 0; e < EDIM; ++e) {
        float x = __shfl(dp, e, 32);
        df = fmaf(x, s_dwT[e * EDIM + lane], df);
    }
    df = fmaxf(df, 0.f);

    float* cb = combined + (size_t)b * CDIM;
    cb[4 + lane]  = hpool;
    cb[36 + lane] = df;
    if (lane < 4) cb[lane] = feats[b * 4 + lane];
}

// ---------------------------------------------------------------------------
// Kernel 2: MLP head via V_WMMA_F32_16X16X4_F32.  One wave per 16-batch tile.
//   A layout (16x4 f32): lane<16 -> M=lane, K={kc,kc+1}; lane>=16 -> K={kc+2,kc+3}
//   B layout (4x16 f32): lane<16 -> K={kc,kc+1}, N=lane; lane>=16 -> K={kc+2,kc+3}
//   C/D (16x16 f32):     c[r] = (M = r + 8*(lane>>4), N = lane&15)
// ---------------------------------------------------------------------------
__global__ __launch_bounds__(32) void k_mlp_head(
    const float* __restrict__ combined,  // [B][68]
    const float* __restrict__ fc1_w,     // [64][68]
    const float* __restrict__ fc1_b,     // [64]
    const float* __restrict__ fc2_w,     // [32][64]
    const float* __restrict__ fc2_b,     // [32]
    const float* __restrict__ out_w,     // [2][32]
    const float* __restrict__ out_b,     // [2]
    float*       __restrict__ out)       // [B][2]
{
    __shared__ float sA [16 * CDIM];   // input tile
    __shared__ float sW1[CDIM * 64];   // fc1_w^T : [k][n]
    __shared__ float sH1[16 * 64];
    __shared__ float sW2[64 * 32];     // fc2_w^T : [k][n]
    __shared__ float sH2[16 * 32];
    __shared__ float sWo[32 * 16];     // out_w^T padded to N=16: [k][n]

    const int lane = threadIdx.x;
    const int b0   = blockIdx.x * 16;
    const int half = lane >> 4;
    const int l16  = lane & 15;

    for (int i = lane; i < 64 * CDIM; i += 32) {
        int n = i / CDIM, kk = i % CDIM;
        sW1[kk * 64 + n] = fc1_w[i];
    }
    for (int i = lane; i < 32 * 64; i += 32) {
        int n = i / 64, kk = i % 64;
        sW2[kk * 32 + n] = fc2_w[i];
    }
    for (int i = lane; i < 32 * 16; i += 32) {
        int kk = i / 16, n = i % 16;
        sWo[i] = (n < 2) ? out_w[n * 32 + kk] : 0.f;
    }
    for (int i = lane; i < 16 * CDIM; i += 32)
        sA[i] = combined[(size_t)b0 * CDIM + i];
    __syncthreads();

    // ---- FC1: 16x68 @ 68x64, ReLU ----
    for (int nt = 0; nt < 4; ++nt) {
        const int nb = nt * 16;
        v8f c;
        float bias = fc1_b[nb + l16];
        #pragma unroll
        for (int r = 0; r < 8; ++r) c[r] = bias;
        #pragma unroll
        for (int kc = 0; kc < CDIM; kc += 4) {
            v2f a, bm;
            a.x  = sA[l16 * CDIM + kc + 2 * half + 0];
            a.y  = sA[l16 * CDIM + kc + 2 * half + 1];
            bm.x = sW1[(kc + 2 * half + 0) * 64 + nb + l16];
            bm.y = sW1[(kc + 2 * half + 1) * 64 + nb + l16];
            c = __builtin_amdgcn_wmma_f32_16x16x4_f32(false, a, false, bm,
                                                      (short)0, c, false, false);
        }
        #pragma unroll
        for (int r = 0; r < 8; ++r)
            sH1[(r + 8 * half) * 64 + nb + l16] = fmaxf(c[r], 0.f);
    }
    __syncthreads();

    // ---- FC2: 16x64 @ 64x32, ReLU ----
    for (int nt = 0; nt < 2; ++nt) {
        const int nb = nt * 16;
        v8f c;
        float bias = fc2_b[nb + l16];
        #pragma unroll
        for (int r = 0; r < 8; ++r) c[r] = bias;
        #pragma unroll
        for (int kc = 0; kc < 64; kc += 4) {
            v2f a, bm;
            a.x  = sH1[l16 * 64 + kc + 2 * half + 0];
            a.y  = sH1[l16 * 64 + kc + 2 * half + 1];
            bm.x = sW2[(kc + 2 * half + 0) * 32 + nb + l16];
            bm.y = sW2[(kc + 2 * half + 1) * 32 + nb + l16];
            c = __builtin_amdgcn_wmma_f32_16x16x4_f32(false, a, false, bm,
                                                      (short)0, c, false, false);
        }
        #pragma unroll
        for (int r = 0; r < 8; ++r)
            sH2[(r + 8 * half) * 32 + nb + l16] = fmaxf(c[r], 0.f);
    }
    __syncthreads();

    // ---- OUT: 16x32 @ 32x16 (N padded, only n<2 valid) ----
    {
        v8f c;
        float bias = (l16 < 2) ? out_b[l16] : 0.f;
        #pragma unroll
        for (int r = 0; r < 8; ++r) c[r] = bias;
        #pragma unroll
        for (int kc = 0; kc < 32; kc += 4) {
            v2f a, bm;
            a.x  = sH2[l16 * 32 + kc + 2 * half + 0];
            a.y  = sH2[l16 * 32 + kc + 2 * half + 1];
            bm.x = sWo[(kc + 2 * half + 0) * 16 + l16];
            bm.y = sWo[(kc + 2 * half + 1) * 16 + l16];
            c = __builtin_amdgcn_wmma_f32_16x16x4_f32(false, a, false, bm,
                                                      (short)0, c, false, false);
        }
        if (l16 < 2) {
            #pragma unroll
            for (int r = 0; r < 8; ++r) {
                int m = r + 8 * half;
                out[(size_t)(b0 + m) * 2 + l16] = c[r];
            }
        }
    }
}

extern "C" void kernel_launch(void* const* d_in, const int* in_sizes, int n_in,
                              void* d_out, int out_size, void* d_ws, size_t ws_size,
                              hipStream_t stream) {
    const float* feats      = (const float*)d_in[0];
    const int*   hand_ids   = (const int*)  d_in[1];
    const int*   deck_ids   = (const int*)  d_in[2];
    const float* embed      = (const float*)d_in[3];
    const float* in_proj_w  = (const float*)d_in[4];
    const float* in_proj_b  = (const float*)d_in[5];
    const float* out_proj_w = (const float*)d_in[6];
    const float* out_proj_b = (const float*)d_in[7];
    const float* ln_g       = (const float*)d_in[8];
    const float* ln_b       = (const float*)d_in[9];
    const float* deck_w     = (const float*)d_in[10];
    const float* deck_b     = (const float*)d_in[11];
    const float* fc1_w      = (const float*)d_in[12];
    const float* fc1_b      = (const float*)d_in[13];
    const float* fc2_w      = (const float*)d_in[14];
    const float* fc2_b      = (const float*)d_in[15];
    const float* out_w      = (const float*)d_in[16];
    const float* out_b      = (const float*)d_in[17];

    float* combined = (float*)d_ws;          // B*68 floats = ~17.8 MB
    float* outp     = (float*)d_out;         // B*2 floats

    k_attn_pool<<<B_TOT / 8, 256, 0, stream>>>(
        feats, hand_ids, deck_ids, embed,
        in_proj_w, in_proj_b, out_proj_w, out_proj_b,
        ln_g, ln_b, deck_w, deck_b, combined);

    k_mlp_head<<<B_TOT / 16, 32, 0, stream>>>(
        combined, fc1_w, fc1_b, fc2_w, fc2_b, out_w, out_b, outp);
}